// TransitionUp_45286135169570
// MI455X (gfx1250) — compile-verified
//
#include <hip/hip_runtime.h>
#include <hip/hip_bf16.h>

// ---------------------------------------------------------------------------
// Problem constants (from reference)
// ---------------------------------------------------------------------------
#define BATCH       4
#define N1_TOTAL    8192          // coarse points
#define N2_TOTAL    32768         // fine points
#define N1_PER_B    (N1_TOTAL / BATCH)   // 2048
#define N2_PER_B    (N2_TOTAL / BATCH)   // 8192
#define FOUT        128
#define KIN1        256
#define KIN2        128
#define BN_EPS      1e-5f

typedef __attribute__((ext_vector_type(2))) float v2f;
typedef __attribute__((ext_vector_type(8))) float v8f;

// ---------------------------------------------------------------------------
// GEMM  Y[M,N] = X[M,K] @ W[K,N] + bias[N]   via V_WMMA_F32_16X16X4_F32
// One wave (32 lanes) computes a 16x64 strip: 4 adjacent 16x16 N-tiles,
// reusing the A fragment 4x per k-step (A traffic cut 4x vs 1 tile/wave).
// Full f32 precision (matches the f32 reference exactly in formulation).
//
// Lane layouts per CDNA5 ISA 7.12.2 (32-bit, wave32):
//   A 16x4  : lane l holds M = l&15,  frag[v] = A[M][2*(l>>4) + v]
//   B 4x16  : lane l holds N = l&15,  frag[v] = B[2*(l>>4) + v][N]
//   C/D     : lane l holds N = l&15,  acc[v]  = D[v + 8*(l>>4)][N]
// ---------------------------------------------------------------------------
__global__ __launch_bounds__(32) void gemm_bias_wmma(
    const float* __restrict__ X, const float* __restrict__ W,
    const float* __restrict__ bias, float* __restrict__ Y,
    int M, int K, int N)
{
    const int strip = blockIdx.x;          // N / 64 strips
    const int tileM = blockIdx.y;          // M / 16 tiles
    const int lane  = threadIdx.x;         // 0..31 (wave32)
    const int half  = lane >> 4;           // 0 / 1
    const int l15   = lane & 15;

    const float* __restrict__ Arow = X + (size_t)(tileM * 16 + l15) * K + half * 2;
    const float* __restrict__ Bbase = W + strip * 64 + l15;

    v8f acc0 = {}, acc1 = {}, acc2 = {}, acc3 = {};
    for (int k0 = 0; k0 < K; k0 += 4) {
        v2f a;
        a.x = Arow[k0 + 0];                         // contiguous pair -> b64
        a.y = Arow[k0 + 1];
        const size_t r0 = (size_t)(k0 + half * 2 + 0) * N;
        const size_t r1 = (size_t)(k0 + half * 2 + 1) * N;
        v2f b0, b1, b2, b3;
        b0.x = Bbase[r0 +  0]; b0.y = Bbase[r1 +  0];
        b1.x = Bbase[r0 + 16]; b1.y = Bbase[r1 + 16];
        b2.x = Bbase[r0 + 32]; b2.y = Bbase[r1 + 32];
        b3.x = Bbase[r0 + 48]; b3.y = Bbase[r1 + 48];
        acc0 = __builtin_amdgcn_wmma_f32_16x16x4_f32(false, a, false, b0,
                                                     (short)0, acc0, false, false);
        acc1 = __builtin_amdgcn_wmma_f32_16x16x4_f32(false, a, false, b1,
                                                     (short)0, acc1, false, false);
        acc2 = __builtin_amdgcn_wmma_f32_16x16x4_f32(false, a, false, b2,
                                                     (short)0, acc2, false, false);
        acc3 = __builtin_amdgcn_wmma_f32_16x16x4_f32(false, a, false, b3,
                                                     (short)0, acc3, false, false);
    }

    const int col0 = strip * 64 + l15;
    const float bv0 = bias[col0 +  0];
    const float bv1 = bias[col0 + 16];
    const float bv2 = bias[col0 + 32];
    const float bv3 = bias[col0 + 48];
#pragma unroll
    for (int v = 0; v < 8; ++v) {
        const size_t row = (size_t)(tileM * 16 + v + 8 * half) * N;
        Y[row + col0 +  0] = acc0[v] + bv0;
        Y[row + col0 + 16] = acc1[v] + bv1;
        Y[row + col0 + 32] = acc2[v] + bv2;
        Y[row + col0 + 48] = acc3[v] + bv3;
    }
}

// ---------------------------------------------------------------------------
// BatchNorm statistics: one block per column, deterministic tree reduction.
// Writes mean[c] and rstd[c] = rsqrt(var + eps).
// ---------------------------------------------------------------------------
__global__ __launch_bounds__(256) void bn_stats(
    const float* __restrict__ Y, int Nrows,
    float* __restrict__ mean, float* __restrict__ rstd)
{
    const int c = blockIdx.x;              // column 0..127
    float s = 0.f, s2 = 0.f;
    for (int r = threadIdx.x; r < Nrows; r += 256) {
        const float v = Y[(size_t)r * FOUT + c];
        s  += v;
        s2 += v * v;
    }
    __shared__ float sh[256];
    __shared__ float sh2[256];
    sh[threadIdx.x] = s; sh2[threadIdx.x] = s2;
    __syncthreads();
    for (int o = 128; o > 0; o >>= 1) {
        if ((int)threadIdx.x < o) {
            sh[threadIdx.x]  += sh[threadIdx.x + o];
            sh2[threadIdx.x] += sh2[threadIdx.x + o];
        }
        __syncthreads();
    }
    if (threadIdx.x == 0) {
        const float m   = sh[0] / (float)Nrows;
        const float var = sh2[0] / (float)Nrows - m * m;
        mean[c] = m;
        rstd[c] = rsqrtf(var + BN_EPS);
    }
}

// ---------------------------------------------------------------------------
// In-place BN affine + ReLU
// ---------------------------------------------------------------------------
__global__ __launch_bounds__(256) void bn_apply_relu(
    float* __restrict__ Y,
    const float* __restrict__ mean, const float* __restrict__ rstd,
    const float* __restrict__ g, const float* __restrict__ be, int total)
{
    const int i = blockIdx.x * 256 + threadIdx.x;
    if (i < total) {
        const int c = i & (FOUT - 1);
        const float v = (Y[i] - mean[c]) * rstd[c];
        Y[i] = fmaxf(g[c] * v + be[c], 0.f);
    }
}

// ---------------------------------------------------------------------------
// KNN phase 1: per fine point, top-3 nearest coarse points (same batch).
// One block = 256 fine points of one batch; coarse positions cached in LDS.
// Writes global coarse indices and pre-normalized inverse-d^2 weights.
// ---------------------------------------------------------------------------
__global__ __launch_bounds__(256) void knn_find(
    const float* __restrict__ p1, const float* __restrict__ p2,
    int* __restrict__ nidx, float* __restrict__ nw)
{
    const int blocksPerBatch = N2_PER_B / 256;          // 32
    const int b     = blockIdx.x / blocksPerBatch;
    const int chunk = blockIdx.x % blocksPerBatch;

    __shared__ float sx[N1_PER_B];
    __shared__ float sy[N1_PER_B];
    __shared__ float sz[N1_PER_B];
    const float* __restrict__ pb = p1 + (size_t)b * N1_PER_B * 3;
    for (int j = threadIdx.x; j < N1_PER_B; j += 256) {
        sx[j] = pb[j * 3 + 0];
        sy[j] = pb[j * 3 + 1];
        sz[j] = pb[j * 3 + 2];
    }
    __syncthreads();

    const int i = b * N2_PER_B + chunk * 256 + threadIdx.x;  // global fine idx
    const float qx = p2[i * 3 + 0];
    const float qy = p2[i * 3 + 1];
    const float qz = p2[i * 3 + 2];

    float d0 = 1e30f, d1 = 1e30f, d2 = 1e30f;
    int   i0 = 0,     i1 = 0,     i2 = 0;
    for (int j = 0; j < N1_PER_B; ++j) {
        const float dx = qx - sx[j];
        const float dy = qy - sy[j];
        const float dz = qz - sz[j];
        const float d  = dx * dx + dy * dy + dz * dz;
        if (d < d2) {
            if (d < d1) {
                if (d < d0) { d2 = d1; i2 = i1; d1 = d0; i1 = i0; d0 = d; i0 = j; }
                else        { d2 = d1; i2 = i1; d1 = d;  i1 = j; }
            } else          { d2 = d;  i2 = j; }
        }
    }

    const float w0 = 1.f / fmaxf(d0, 1e-16f);
    const float w1 = 1.f / fmaxf(d1, 1e-16f);
    const float w2 = 1.f / fmaxf(d2, 1e-16f);
    const float inv = 1.f / (w0 + w1 + w2);
    nidx[i * 3 + 0] = b * N1_PER_B + i0;
    nidx[i * 3 + 1] = b * N1_PER_B + i1;
    nidx[i * 3 + 2] = b * N1_PER_B + i2;
    nw[i * 3 + 0] = w0 * inv;
    nw[i * 3 + 1] = w1 * inv;
    nw[i * 3 + 2] = w2 * inv;
}

// ---------------------------------------------------------------------------
// KNN phase 2: out[i,:] += sum_k w_k * f1[idx_k,:]   (out already holds f2act)
// One block per fine point, 128 threads = 128 features (fully coalesced).
// ---------------------------------------------------------------------------
__global__ __launch_bounds__(128) void interp_add(
    const float* __restrict__ f1, const int* __restrict__ nidx,
    const float* __restrict__ nw, float* __restrict__ out)
{
    const int i = blockIdx.x;
    const int f = threadIdx.x;
    const int   j0 = nidx[i * 3 + 0], j1 = nidx[i * 3 + 1], j2 = nidx[i * 3 + 2];
    const float w0 = nw[i * 3 + 0],   w1 = nw[i * 3 + 1],   w2 = nw[i * 3 + 2];
    out[(size_t)i * FOUT + f] += w0 * f1[(size_t)j0 * FOUT + f]
                               + w1 * f1[(size_t)j1 * FOUT + f]
                               + w2 * f1[(size_t)j2 * FOUT + f];
}

// ---------------------------------------------------------------------------
// Tail: copy positions_2 and batch_2 (int -> float) into output tuple tail.
// ---------------------------------------------------------------------------
__global__ __launch_bounds__(256) void tail_copy(
    const float* __restrict__ pos2, const int* __restrict__ batch2,
    float* __restrict__ out)
{
    const int i = blockIdx.x * 256 + threadIdx.x;
    const int P = N2_TOTAL * 3;
    float* tail = out + (size_t)N2_TOTAL * FOUT;
    if (i < P)                 tail[i] = pos2[i];
    else if (i < P + N2_TOTAL) tail[i] = (float)batch2[i - P];
}

// ---------------------------------------------------------------------------
// Host-side launcher
// ---------------------------------------------------------------------------
extern "C" void kernel_launch(void* const* d_in, const int* in_sizes, int n_in,
                              void* d_out, int out_size, void* d_ws, size_t ws_size,
                              hipStream_t stream) {
    (void)in_sizes; (void)n_in; (void)out_size; (void)ws_size;

    const float* features_1  = (const float*)d_in[0];   // [8192,256]
    const float* positions_1 = (const float*)d_in[1];   // [8192,3]
    // d_in[2] = batch_1 (unused: batches are fixed equal segments)
    const float* features_2  = (const float*)d_in[3];   // [32768,128]
    const float* positions_2 = (const float*)d_in[4];   // [32768,3]
    const int*   batch_2     = (const int*)  d_in[5];   // [32768]
    const float* W1  = (const float*)d_in[6];           // [256,128]
    const float* b1  = (const float*)d_in[7];
    const float* g1  = (const float*)d_in[8];
    const float* be1 = (const float*)d_in[9];
    const float* W2  = (const float*)d_in[10];          // [128,128]
    const float* b2  = (const float*)d_in[11];
    const float* g2  = (const float*)d_in[12];
    const float* be2 = (const float*)d_in[13];

    float* out = (float*)d_out;

    // Workspace layout
    char* ws = (char*)d_ws;
    float* f1a   = (float*)ws;                                   // 8192*128 f32
    ws += (size_t)N1_TOTAL * FOUT * sizeof(float);
    float* mean1 = (float*)ws; ws += FOUT * sizeof(float);
    float* rstd1 = (float*)ws; ws += FOUT * sizeof(float);
    float* mean2 = (float*)ws; ws += FOUT * sizeof(float);
    float* rstd2 = (float*)ws; ws += FOUT * sizeof(float);
    int*   nidx  = (int*)ws;   ws += (size_t)N2_TOTAL * 3 * sizeof(int);
    float* nw    = (float*)ws; ws += (size_t)N2_TOTAL * 3 * sizeof(float);

    // 1) GEMM1: y1 = features_1 @ W1 + b1  -> f1a   (WMMA f32, 16x64 strips)
    gemm_bias_wmma<<<dim3(FOUT / 64, N1_TOTAL / 16), 32, 0, stream>>>(
        features_1, W1, b1, f1a, N1_TOTAL, KIN1, FOUT);

    // 2) GEMM2: y2 = features_2 @ W2 + b2  -> d_out[0 : 32768*128]
    gemm_bias_wmma<<<dim3(FOUT / 64, N2_TOTAL / 16), 32, 0, stream>>>(
        features_2, W2, b2, out, N2_TOTAL, KIN2, FOUT);

    // 3) BN stats (deterministic per-column reductions)
    bn_stats<<<FOUT, 256, 0, stream>>>(f1a, N1_TOTAL, mean1, rstd1);
    bn_stats<<<FOUT, 256, 0, stream>>>(out,  N2_TOTAL, mean2, rstd2);

    // 4) BN affine + ReLU, in place
    bn_apply_relu<<<(N1_TOTAL * FOUT) / 256, 256, 0, stream>>>(
        f1a, mean1, rstd1, g1, be1, N1_TOTAL * FOUT);
    bn_apply_relu<<<(N2_TOTAL * FOUT) / 256, 256, 0, stream>>>(
        out, mean2, rstd2, g2, be2, N2_TOTAL * FOUT);

    // 5) KNN top-3 (LDS-cached coarse positions)
    knn_find<<<(N2_TOTAL / 256), 256, 0, stream>>>(
        positions_1, positions_2, nidx, nw);

    // 6) interp-and-add into d_out
    interp_add<<<N2_TOTAL, 128, 0, stream>>>(f1a, nidx, nw, out);

    // 7) tuple tail: positions_2 then batch_2 (as float)
    tail_copy<<<(N2_TOTAL * 3 + N2_TOTAL + 255) / 256, 256, 0, stream>>>(
        positions_2, batch_2, out);
}